// GeometryEncoder_12549894439644
// MI455X (gfx1250) — compile-verified
//
#include <hip/hip_runtime.h>
#include <math.h>

// ---------- types for WMMA f32 16x16x4 ----------
typedef __attribute__((ext_vector_type(2))) float v2f;
typedef __attribute__((ext_vector_type(8))) float v8f;

#define Bn   16
#define Nn   256
#define DIMn 256

// workspace layout (floats):
//   part : [Bn][16][96]  per-block partial hidden sums   (24576 floats)
//   emb  : [Bn][160]                                      (2560 floats)
#define WS_PART 0
#define WS_EMB  (Bn * 16 * 96)

__device__ __forceinline__ float gelu_exact(float x) {
    // gelu(x) = 0.5*x*(1+erf(x/sqrt(2))), approximate=False
    return 0.5f * x * (1.0f + erff(x * 0.70710678118654752440f));
}

// -----------------------------------------------------------------------------
// Kernel 1: masked pair-sum of first-layer hidden activations.
// grid = (B, 16), block = 256 (8 waves). Each block: 16 i-rows of one batch.
// lane == hidden dim k (0..31); pair scalars broadcast via __shfl.
// Writes deterministic per-block partials: part[(b*16+by)*96 + t].
// -----------------------------------------------------------------------------
__global__ __launch_bounds__(256) void k1_pairs(
    const float* __restrict__ pos, const int* __restrict__ mask,
    const float* __restrict__ dw1, const float* __restrict__ db1,
    const float* __restrict__ aw1, const float* __restrict__ ab1,
    const float* __restrict__ hw1, const float* __restrict__ hb1,
    float* __restrict__ part)
{
    __shared__ float swave[8 * 96];

    const int t    = threadIdx.x;
    const int lane = t & 31;
    const int wv   = t >> 5;
    const int b    = blockIdx.x;

    // per-lane (== per hidden dim) first-layer weights, loop-invariant
    const float wd  = dw1[lane];
    const float bd  = db1[lane];
    const float wa0 = aw1[lane];        // aw1[0][k]
    const float wa1 = aw1[32 + lane];   // aw1[1][k]
    const float ba  = ab1[lane];
    const float wh  = hw1[lane];
    const float bh  = hb1[lane];

    const float* posb  = pos  + (size_t)b * Nn * 3;
    const int*   maskb = mask + (size_t)b * Nn;

    float accd = 0.0f, acca = 0.0f, acch = 0.0f;

    for (int rr = 0; rr < 2; ++rr) {
        const int i   = blockIdx.y * 16 + wv * 2 + rr;
        const float pix = posb[i * 3 + 0];
        const float piy = posb[i * 3 + 1];
        const float piz = posb[i * 3 + 2];
        const float mi  = (maskb[i] != 0) ? 1.0f : 0.0f;

        for (int jg = 0; jg < 8; ++jg) {
            const int j = jg * 32 + lane;
            // delta[b,i,j] = pos[b,j] - pos[b,i]
            const float dx = posb[j * 3 + 0] - pix;
            const float dy = posb[j * 3 + 1] - piy;
            const float dz = posb[j * 3 + 2] - piz;
            const float pm = mi * ((maskb[j] != 0) ? 1.0f : 0.0f);

            const float sq   = dx * dx + dy * dy + dz * dz;
            const float dist = (sq > 0.0f) ? sqrtf(sq) : 0.0f;
            const float r2   = dx * dx + dy * dy;
            const bool  zr   = (dx == 0.0f) && (dy == 0.0f);
            const float rxy  = zr ? 1.0f : sqrtf(r2);
            const float cs   = zr ? 1.0f : (dx / rxy);   // cos(atan2(dy,dx))
            const float sn   = zr ? 0.0f : (dy / rxy);   // sin(atan2(dy,dx))

            // broadcast each pair p; all 32 lanes evaluate hidden dim k=lane
            for (int p = 0; p < 32; ++p) {
                const float dist_p = __shfl(dist, p, 32);
                const float cs_p   = __shfl(cs,   p, 32);
                const float sn_p   = __shfl(sn,   p, 32);
                const float alt_p  = __shfl(dz,   p, 32);
                const float pm_p   = __shfl(pm,   p, 32);

                accd += pm_p * gelu_exact(fmaf(dist_p, wd, bd));
                acca += pm_p * gelu_exact(fmaf(sn_p, wa1, fmaf(cs_p, wa0, ba)));
                acch += pm_p * gelu_exact(fmaf(alt_p, wh, bh));
            }
        }
    }

    // deterministic in-block reduction (unique LDS slots, no atomics)
    swave[wv * 96 + lane]      = accd;
    swave[wv * 96 + 32 + lane] = acca;
    swave[wv * 96 + 64 + lane] = acch;
    __syncthreads();

    if (t < 96) {
        float s = 0.0f;
        #pragma unroll
        for (int w2 = 0; w2 < 8; ++w2) s += swave[w2 * 96 + t];
        part[((size_t)b * 16 + blockIdx.y) * 96 + t] = s;
    }
}

// -----------------------------------------------------------------------------
// Kernel 2: reduce partials -> hsum, apply second-layer matmuls:
//   emb[b][0:64]    = (hsum_d/cnt) @ dw2 + db2
//   emb[b][64:128]  = (hsum_a/cnt) @ aw2 + ab2
//   emb[b][128:160] = (hsum_h/cnt) @ hw2 + hb2
// cnt = max((sum mask)^2, 1); bias zeroed if sum(pm)==0 (reference semantics).
// one block, 256 threads.
// -----------------------------------------------------------------------------
__global__ __launch_bounds__(256) void k2_emb(
    const int* __restrict__ mask, const float* __restrict__ part,
    const float* __restrict__ dw2, const float* __restrict__ db2,
    const float* __restrict__ aw2, const float* __restrict__ ab2,
    const float* __restrict__ hw2, const float* __restrict__ hb2,
    float* __restrict__ emb)
{
    __shared__ float s_hsum[Bn * 96];
    __shared__ float s_invc[Bn];
    __shared__ float s_bsc[Bn];

    const int t = threadIdx.x;

    for (int i = t; i < Bn * 96; i += 256) {
        const int b = i / 96, k = i % 96;
        float s = 0.0f;
        #pragma unroll
        for (int blk = 0; blk < 16; ++blk)
            s += part[((size_t)b * 16 + blk) * 96 + k];
        s_hsum[i] = s;
    }
    if (t < Bn) {
        int m = 0;
        for (int j = 0; j < Nn; ++j) m += (mask[t * Nn + j] != 0);
        const float cnt = (float)m * (float)m;       // sum(pm) = (sum mask)^2
        s_bsc[t]  = (cnt > 0.0f) ? 1.0f : 0.0f;
        s_invc[t] = 1.0f / fmaxf(cnt, 1.0f);
    }
    __syncthreads();

    for (int it = 0; it < 10; ++it) {                // 16*160 = 2560 outputs
        const int idx = it * 256 + t;
        const int b = idx / 160, f = idx % 160;
        const float invc = s_invc[b];
        const float bsc  = s_bsc[b];
        const float* hs  = &s_hsum[b * 96];
        float acc;
        if (f < 64) {
            acc = bsc * db2[f];
            #pragma unroll
            for (int k = 0; k < 32; ++k) acc = fmaf(hs[k] * invc, dw2[k * 64 + f], acc);
        } else if (f < 128) {
            const int fa = f - 64;
            acc = bsc * ab2[fa];
            #pragma unroll
            for (int k = 0; k < 32; ++k) acc = fmaf(hs[32 + k] * invc, aw2[k * 64 + fa], acc);
        } else {
            const int fh = f - 128;
            acc = bsc * hb2[fh];
            #pragma unroll
            for (int k = 0; k < 32; ++k) acc = fmaf(hs[64 + k] * invc, hw2[k * 32 + fh], acc);
        }
        emb[idx] = acc;
    }
}

// -----------------------------------------------------------------------------
// Kernel 3: x = emb(16x160) @ gw(160x256) + gb via V_WMMA_F32_16X16X4_F32,
// then fused LayerNorm (population var) + exact gelu.
// one block, 512 threads = 16 waves; wave w owns N-columns [16w, 16w+16).
// -----------------------------------------------------------------------------
#define ESTR 161   // padded LDS row stride (odd -> bank-conflict-free)

__global__ __launch_bounds__(512) void k3_gemm_ln(
    const float* __restrict__ emb, const float* __restrict__ gw,
    const float* __restrict__ gb,  const float* __restrict__ ln_g,
    const float* __restrict__ ln_b, float* __restrict__ out)
{
    __shared__ float se[16 * ESTR];
    __shared__ float rs[16][16];   // [row][wave] partial sums
    __shared__ float rq[16][16];   // [row][wave] partial sums of squares
    __shared__ float s_mu[16];
    __shared__ float s_rv[16];

    const int t = threadIdx.x;

    // stage emb into LDS (2560 = 512*5, fully uniform)
    for (int i = t; i < Bn * 160; i += 512) {
        const int r = i / 160, c = i % 160;
        se[r * ESTR + c] = emb[i];
    }
    __syncthreads();

    const int w    = t >> 5;       // wave id == N tile
    const int lane = t & 31;
    const int half = lane >> 4;    // 0: K{0,1}/M0-7 rows ; 1: K{2,3}/M8-15 rows
    const int lr   = lane & 15;
    const int col  = w * 16 + lr;

    v8f acc = {};
    for (int k0 = 0; k0 < 160; k0 += 4) {
        const int kk = k0 + 2 * half;
        v2f a, bv;
        // A 16x4 f32 fragment: lanes 0-15 -> K=k0,k0+1 ; lanes 16-31 -> K=k0+2,k0+3
        a.x = se[lr * ESTR + kk];
        a.y = se[lr * ESTR + kk + 1];
        // B 4x16 f32 fragment: row K striped across lanes (N = lr per half)
        bv.x = gw[kk * DIMn + col];
        bv.y = gw[(kk + 1) * DIMn + col];
        acc = __builtin_amdgcn_wmma_f32_16x16x4_f32(
            false, a, false, bv, (short)0, acc, false, false);
    }

    // add gb, compute per-row partial sums for LayerNorm
    const float gbc = gb[col];
    float x[8];
    #pragma unroll
    for (int r = 0; r < 8; ++r) {
        x[r] = acc[r] + gbc;                 // element (row = r + 8*half, col)
        float v  = x[r];
        float vq = v * v;
        #pragma unroll
        for (int off = 1; off < 16; off <<= 1) {   // reduce within 16-lane half
            v  += __shfl_xor(v,  off, 32);
            vq += __shfl_xor(vq, off, 32);
        }
        if (lr == 0) {
            const int row = r + 8 * half;
            rs[row][w] = v;
            rq[row][w] = vq;
        }
    }
    __syncthreads();

    if (t < 16) {
        float s = 0.0f, q = 0.0f;
        #pragma unroll
        for (int w2 = 0; w2 < 16; ++w2) { s += rs[t][w2]; q += rq[t][w2]; }
        const float mu  = s * (1.0f / DIMn);
        const float var = q * (1.0f / DIMn) - mu * mu;
        s_mu[t] = mu;
        s_rv[t] = rsqrtf(var + 1e-5f);
    }
    __syncthreads();

    const float lg = ln_g[col];
    const float lb = ln_b[col];
    #pragma unroll
    for (int r = 0; r < 8; ++r) {
        const int row = r + 8 * half;
        const float xn = (x[r] - s_mu[row]) * s_rv[row] * lg + lb;
        out[row * DIMn + col] = gelu_exact(xn);
    }
}

// -----------------------------------------------------------------------------
extern "C" void kernel_launch(void* const* d_in, const int* in_sizes, int n_in,
                              void* d_out, int out_size, void* d_ws, size_t ws_size,
                              hipStream_t stream) {
    const float* positions = (const float*)d_in[0];
    const int*   mask      = (const int*)  d_in[1];
    const float* dw1 = (const float*)d_in[2];
    const float* db1 = (const float*)d_in[3];
    const float* dw2 = (const float*)d_in[4];
    const float* db2 = (const float*)d_in[5];
    const float* aw1 = (const float*)d_in[6];
    const float* ab1 = (const float*)d_in[7];
    const float* aw2 = (const float*)d_in[8];
    const float* ab2 = (const float*)d_in[9];
    const float* hw1 = (const float*)d_in[10];
    const float* hb1 = (const float*)d_in[11];
    const float* hw2 = (const float*)d_in[12];
    const float* hb2 = (const float*)d_in[13];
    const float* gw  = (const float*)d_in[14];
    const float* gb  = (const float*)d_in[15];
    const float* ln_g = (const float*)d_in[16];
    const float* ln_b = (const float*)d_in[17];

    float* wsf  = (float*)d_ws;
    float* part = wsf + WS_PART;   // 16*16*96 floats
    float* emb  = wsf + WS_EMB;    // 16*160  floats
    float* outf = (float*)d_out;   // 16*256  f32

    k1_pairs<<<dim3(Bn, 16), 256, 0, stream>>>(positions, mask,
                                               dw1, db1, aw1, ab1, hw1, hb1, part);
    k2_emb<<<1, 256, 0, stream>>>(mask, part, dw2, db2, aw2, ab2, hw2, hb2, emb);
    k3_gemm_ln<<<1, 512, 0, stream>>>(emb, gw, gb, ln_g, ln_b, outf);
}